// LocalConvolutionDilate_75325136437780
// MI455X (gfx1250) — compile-verified
//
#include <hip/hip_runtime.h>

// Problem constants (from reference)
#define B_   8
#define C_   256
#define H_   56
#define W_   56
#define WC_  32
#define G_   8            // C_ / WC_ channels sharing one weight channel
#define K_   9
#define TH_  4            // rows per block tile
#define HW_  (H_ * W_)    // 3136

typedef unsigned int u32x4 __attribute__((ext_vector_type(4)));
typedef int          i32x8 __attribute__((ext_vector_type(8)));
typedef int          i32x4 __attribute__((ext_vector_type(4)));

__global__ __launch_bounds__(W_ * TH_)
void lcd_tdm_kernel(const float* __restrict__ x,
                    const float* __restrict__ weight,
                    const int*   __restrict__ dilation,
                    float*       __restrict__ out)
{
    // Dynamic LDS starts at LDS offset 0 -> descriptor lds_addr = 0.
    extern __shared__ float sw[];          // [K_][TH_][W_] = 9*4*56 floats = 8064 B

    const int wc = blockIdx.x;             // weight channel 0..31
    const int h0 = blockIdx.y * TH_;       // row tile base
    const int b  = blockIdx.z;             // batch

    const int wv = threadIdx.x;            // 0..55  (w)
    const int hl = threadIdx.y;            // 0..3   (row within tile)
    const int h  = h0 + hl;

    // ---- Wave 0: issue one 3D TDM load of the weight tile into LDS ----
    // weight[b,0,wc,k,h,w] flat = ((b*32 + wc)*9 + k)*3136 + h*56 + w
    if (threadIdx.y == 0 && threadIdx.x < 32) {
        unsigned long long ga = (unsigned long long)(uintptr_t)
            (weight + ((size_t)(b * WC_ + wc) * K_) * HW_ + (size_t)h0 * W_);

        u32x4 g0;
        g0[0] = 1u;                                   // count=1 (valid user D#)
        g0[1] = 0u;                                   // lds_addr = 0
        g0[2] = (unsigned)(ga & 0xFFFFFFFFull);       // global_addr[31:0]
        g0[3] = (unsigned)(ga >> 32) | (2u << 30);    // global_addr[56:32] | type=2

        i32x8 g1;
        g1[0] = 2 << 16;                 // data_size = 4 bytes
        g1[1] = W_ << 16;                // tensor_dim0 = 56
        g1[2] = TH_ << 16;               // tensor_dim1 = TH_
        g1[3] = W_ << 16;                // tile_dim0 = 56 (w, contiguous)
        g1[4] = TH_ | (K_ << 16);        // tile_dim1 = TH_ (h), tile_dim2 = 9 (k)
        g1[5] = W_;                      // tensor_dim0_stride = 56 (h stride)
        g1[6] = (HW_ & 0xFFFF) << 16;    // tensor_dim1_stride lo = 3136 (k stride)
        g1[7] = HW_ >> 16;               // tensor_dim1_stride hi = 0

        i32x4 g2; g2[0] = K_; g2[1] = 0; g2[2] = 0; g2[3] = 0;   // tensor_dim2 = 9
        i32x4 g3; g3[0] = 0;  g3[1] = 0; g3[2] = 0; g3[3] = 0;
        i32x8 g4; g4[0] = 0;  g4[1] = 0; g4[2] = 0; g4[3] = 0;   // extra payload
        g4[4] = 0; g4[5] = 0; g4[6] = 0; g4[7] = 0;              // (clang-23 form)

        __builtin_amdgcn_tensor_load_to_lds(g0, g1, g2, g3, g4, 0);
    }

    // ---- All threads: tap offsets + padding masks (independent of LDS) ----
    const int d = dilation[wc];            // uniform per block -> scalar load

    int   off[K_];
    float msk[K_];
    #pragma unroll
    for (int k = 0; k < K_; ++k) {
        const int hh = h  + (k / 3 - 1) * d;
        const int ww = wv + (k % 3 - 1) * d;
        const bool v = (hh >= 0) & (hh < H_) & (ww >= 0) & (ww < W_);
        off[k] = v ? (hh * W_ + ww) : 0;   // safe in-bounds dummy when padded
        msk[k] = v ? 1.0f : 0.0f;
    }

    // ---- Wait for TDM completion, then make LDS visible to all waves ----
    if (threadIdx.y == 0 && threadIdx.x < 32) {
        __builtin_amdgcn_s_wait_tensorcnt(0);
    }
    __syncthreads();

    // Per-position weights are shared by all 8 channels in the group:
    // load once from LDS, fold in the zero-padding masks.
    float wk[K_];
    #pragma unroll
    for (int k = 0; k < K_; ++k)
        wk[k] = sw[(k * TH_ + hl) * W_ + wv] * msk[k];

    const float* xb = x   + ((size_t)b * C_ + wc) * HW_;
    float*       ob = out + ((size_t)b * C_ + wc) * HW_ + (size_t)h * W_ + wv;

    #pragma unroll
    for (int g = 0; g < G_; ++g) {
        const float* xc = xb + (size_t)g * (WC_ * HW_);
        float acc = 0.0f;
        #pragma unroll
        for (int k = 0; k < K_; ++k)
            acc = fmaf(wk[k], xc[off[k]], acc);
        ob[(size_t)g * (WC_ * HW_)] = acc;
    }
}

extern "C" void kernel_launch(void* const* d_in, const int* in_sizes, int n_in,
                              void* d_out, int out_size, void* d_ws, size_t ws_size,
                              hipStream_t stream) {
    const float* x   = (const float*)d_in[0];
    const float* w   = (const float*)d_in[1];
    const int*   dil = (const int*)d_in[2];
    float*       out = (float*)d_out;

    dim3 grid(WC_, H_ / TH_, B_);          // 32 x 14 x 8 = 3584 blocks
    dim3 block(W_, TH_, 1);                // 224 threads = 7 waves
    size_t shmem = (size_t)K_ * TH_ * W_ * sizeof(float);   // 8064 B

    lcd_tdm_kernel<<<grid, block, shmem, stream>>>(x, w, dil, out);
}